// Agent_56822417326533
// MI455X (gfx1250) — compile-verified
//
#include <hip/hip_runtime.h>
#include <stdint.h>

typedef unsigned long long u64;
typedef __attribute__((ext_vector_type(4))) float vf4;   // native clang vector: OK for NT builtins

#define LSB8 0x0101010101010101ULL

// Column-validity masks for masked horizontal/diagonal shifts.
// mR(k): destination cells (r,c) with c+k <= 7 valid.  mL(k): c-k >= 0 valid.
__device__ __forceinline__ u64 mR(int k){ return LSB8 * (0xFFULL >> k); }
__device__ __forceinline__ u64 mL(int k){ return LSB8 * ((0xFFULL << k) & 0xFFULL); }

// fsh: bit(r,c) := m(r + k*di, c + k*dj)   (0 outside the board)
// bsh: bit(r,c) := m(r - k*di, c - k*dj)
// dir: 0=(0,1) horizontal, 1=(1,0) vertical, 2=(1,1) diag, 3=(1,-1) anti-diag
__device__ __forceinline__ u64 fsh(u64 m, int dir, int k){
  switch(dir){
    case 0:  return (m >> k)     & mR(k);
    case 1:  return  m >> (8*k);
    case 2:  return (m >> (9*k)) & mR(k);
    default: return (m >> (7*k)) & mL(k);
  }
}
__device__ __forceinline__ u64 bsh(u64 m, int dir, int k){
  switch(dir){
    case 0:  return (m << k)     & mL(k);
    case 1:  return  m << (8*k);
    case 2:  return (m << (9*k)) & mL(k);
    default: return (m << (7*k)) & mR(k);
  }
}

// Exact run-length == 1/2/3 masks, OR-ed over the 4 directions.
__device__ __forceinline__ void runChannels(u64 m, u64& c1, u64& c2, u64& c3){
  c1 = 0; c2 = 0; c3 = 0;
#pragma unroll
  for(int d = 0; d < 4; ++d){
    u64 n1 = fsh(m,d,1), n2 = fsh(m,d,2), n3 = fsh(m,d,3);
    u64 p1 = bsh(m,d,1), p2 = bsh(m,d,2), p3 = bsh(m,d,3);
    c1 |= m & ~p1 & ~n1;
    c2 |= m & ((p1 & ~p2 & ~n1) | (n1 & ~n2 & ~p1));
    c3 |= m & ((n1 & n2 & ~n3 & ~p1) |
               (p1 & n1 & ~p2 & ~n2) |
               (p1 & p2 & ~p3 & ~n1));
  }
}

// Line features along rows (DIR=0) or columns (DIR=1); OR into l2/l3/r3.
// Patterns follow the reference's _line_features with bitboard windows; the
// masked forward shifts zero out-of-line windows, enforcing k-range bounds.
template<int DIR>
__device__ __forceinline__ void lineFeats(u64 me, u64 op, u64 empty,
                                          u64& l2, u64& l3, u64& r3){
  const u64 E0  = (DIR == 0) ? LSB8               : 0xFFULL;               // line index 0
  const u64 EHI = (DIR == 0) ? (LSB8 * 0xF8ULL)   : 0xFFFFFFFFFF000000ULL; // index >= 3 (idx+5 OOB)
  u64 nm = ~me;
  u64 f1me = fsh(me,DIR,1), f2me = fsh(me,DIR,2), f3me = fsh(me,DIR,3);
  u64 f2em = fsh(empty,DIR,2), f3em = fsh(empty,DIR,3), f4em = fsh(empty,DIR,4);
  u64 f4nm = fsh(nm,DIR,4);

  // a: X X . .  (both gaps empty) -> mark k,k+1
  u64 A = me & f1me & f2em & f3em;
  l2 |= A | bsh(A,DIR,1);
  // b: . X X .  -> mark k+1,k+2
  u64 Bp = empty & f1me & f2me & f3em;
  l2 |= bsh(Bp,DIR,1) | bsh(Bp,DIR,2);

  u64 core3 = f1me & f2me & f3me;
  // c: . X X X .  (ends empty) -> live3 marks k+1..k+3
  u64 C = empty & core3 & f4em;
  l3 |= bsh(C,DIR,1) | bsh(C,DIR,2) | bsh(C,DIR,3);

  // blocked(k-1) / blocked(k+5): opponent stone or board edge
  u64 lb = E0  | bsh(op,DIR,1);
  u64 rb = EHI | fsh(op,DIR,5);

  // d: [blocked] X X X empty notme -> rush3 marks k..k+2
  u64 D = me & f1me & f2me & f3em & f4nm & lb;
  r3 |= D | bsh(D,DIR,1) | bsh(D,DIR,2);

  // e: open3 (notme XXX notme) with exactly one side blocked -> marks k+1..k+3
  u64 O3 = nm & core3 & f4nm;
  u64 Ep = O3 & (lb ^ rb);
  r3 |= bsh(Ep,DIR,1) | bsh(Ep,DIR,2) | bsh(Ep,DIR,3);
}

__device__ __forceinline__ void async_b128_to_lds(unsigned ldsOff, const void* src){
  asm volatile("global_load_async_to_lds_b128 %0, %1, off"
               :: "v"(ldsOff), "v"(src) : "memory");
}

__global__ __launch_bounds__(128)
void Agent_gomoku_feat_kernel(const float* __restrict__ state,
                              const float* __restrict__ side,
                              float* __restrict__ out, int nB)
{
  // 128 boards per block. Stage: 272B/board (68 floats, 16B-aligned pad ->
  // at most 2-way LDS bank conflicts on the per-thread board read).
  __shared__ __align__(16) float stage[128 * 68];     // 34,816 B
  __shared__ u64 masks[128][18];                      // 18,432 B

  const int t  = threadIdx.x;
  const int b0 = blockIdx.x * 128;

  // ---------- Phase 1: async global -> LDS staging (CDNA5 async engine) ----
  {
    const char* gbase    = (const char*)state + (size_t)b0 * 256; // 256 B/board
    unsigned    ldsStage = (unsigned)(size_t)(&stage[0]);         // LDS byte offset
    if (b0 + 128 <= nB) {
      // Block-uniform fast path: no per-chunk guards, 16 back-to-back issues.
#pragma unroll
      for(int k = 0; k < 16; ++k){
        int c   = t + k * 128;        // 16B chunk id within block (2048 total)
        int brd = c >> 4, j = c & 15; // board, chunk-in-board
        async_b128_to_lds(ldsStage + (unsigned)(brd * 272 + j * 16),
                          gbase + (size_t)brd * 256 + j * 16);
      }
    } else {
#pragma unroll
      for(int k = 0; k < 16; ++k){
        int c   = t + k * 128;
        int brd = c >> 4, j = c & 15;
        if (b0 + brd < nB)
          async_b128_to_lds(ldsStage + (unsigned)(brd * 272 + j * 16),
                            gbase + (size_t)brd * 256 + j * 16);
      }
    }
#if __has_builtin(__builtin_amdgcn_s_wait_asynccnt)
    __builtin_amdgcn_s_wait_asynccnt(0);
#else
    asm volatile("s_wait_asynccnt 0" ::: "memory");
#endif
    __syncthreads();
  }

  // ---------- Phase 2: build bitboards, compute all 18 channel masks -------
  const int  b     = b0 + t;
  const bool valid = (b < nB);
  u64 my = 0, op = 0;
  if (valid){
    const float s   = side[b];
    const vf4*  row = (const vf4*)(&stage[t * 68]);
#pragma unroll
    for(int q = 0; q < 16; ++q){
      vf4 v = row[q];
      int base = 4 * q;
      my |= ((u64)(v.x ==  s) << base)     | ((u64)(v.y ==  s) << (base+1)) |
            ((u64)(v.z ==  s) << (base+2)) | ((u64)(v.w ==  s) << (base+3));
      op |= ((u64)(v.x == -s) << base)     | ((u64)(v.y == -s) << (base+1)) |
            ((u64)(v.z == -s) << (base+2)) | ((u64)(v.w == -s) << (base+3));
    }
  }
  u64 empty = ~(my | op);

  u64 m1, m2, m3, o1, o2, o3;
  runChannels(my, m1, m2, m3);
  runChannels(op, o1, o2, o3);

  u64 l2m = 0, l3m = 0, r3m = 0;
  lineFeats<0>(my, op, empty, l2m, l3m, r3m);
  lineFeats<1>(my, op, empty, l2m, l3m, r3m);
  u64 l2o = 0, l3o = 0, r3o = 0;
  lineFeats<0>(op, my, empty, l2o, l3o, r3o);
  lineFeats<1>(op, my, empty, l2o, l3o, r3o);

  u64* mw = masks[t];
  mw[0]  = my;  mw[1]  = op;
  mw[2]  = m1;  mw[3]  = m2;  mw[4]  = m3;
  mw[5]  = o1;  mw[6]  = o2;  mw[7]  = o3;
  mw[8]  = l2m; mw[9]  = l3m; mw[10] = r3m;
  mw[11] = (__popcll(l2m) >= 2)                 ? ~0ULL : 0ULL;
  mw[12] = (__popcll(l3m) + __popcll(r3m) >= 2) ? ~0ULL : 0ULL;
  mw[13] = l2o; mw[14] = l3o; mw[15] = r3o;
  mw[16] = (__popcll(l2o) >= 2)                 ? ~0ULL : 0ULL;
  mw[17] = (__popcll(l3o) + __popcll(r3o) >= 2) ? ~0ULL : 0ULL;
  __syncthreads();

  // ---------- Phase 3: coalesced non-temporal expansion to float32 ---------
  // Block writes floats [b0*1152, (b0+128)*1152). float4 q = t + k*128 covers
  // float offset f = 4t + 512k; consecutive lanes -> consecutive 16B stores.
  // Output is write-once / never re-read: stream past L2 with NT stores.
  vf4*      outp  = (vf4*)out + (size_t)b0 * 288; // 288 vec4 per board
  int       rem   = 4 * t;     // f - board*1152, < 1152
  int       board = 0;
  const int bit   = rem & 63;  // constant: 512 and 1152 are multiples of 64
#pragma unroll 4
  for(int k = 0; k < 288; ++k){
    int chn = rem >> 6;                       // 0..17
    u64 m   = masks[board][chn];              // near-broadcast across lanes
    unsigned nib = (unsigned)((m >> bit) & 0xFULL);
    vf4 o;
    o.x = (nib & 1u) ? 1.0f : 0.0f;
    o.y = (nib & 2u) ? 1.0f : 0.0f;
    o.z = (nib & 4u) ? 1.0f : 0.0f;
    o.w = (nib & 8u) ? 1.0f : 0.0f;
    if (b0 + board < nB)
      __builtin_nontemporal_store(o, &outp[t + k * 128]);
    rem += 512;
    if (rem >= 1152){ rem -= 1152; ++board; }
  }
}

extern "C" void kernel_launch(void* const* d_in, const int* in_sizes, int n_in,
                              void* d_out, int out_size, void* d_ws, size_t ws_size,
                              hipStream_t stream)
{
  const float* state = (const float*)d_in[0]; // (B,8,8) float32
  const float* side  = (const float*)d_in[1]; // (B,)    float32
  float*       out   = (float*)d_out;         // (B,18,8,8) float32

  int nB     = in_sizes[0] / 64;
  int blocks = (nB + 127) / 128;
  hipLaunchKernelGGL(Agent_gomoku_feat_kernel, dim3(blocks), dim3(128), 0, stream,
                     state, side, out, nB);
}